// GINModel2Layers_67482526155420
// MI455X (gfx1250) — compile-verified
//
#include <hip/hip_runtime.h>

typedef __attribute__((ext_vector_type(2))) float v2f;
typedef __attribute__((ext_vector_type(8))) float v8f;

#define NN 20000
#define NE 600000
#define HID 128
#define LDSS 132  // padded LDS row stride: bank = (4*row + col) % 64, conflict-free

// ---------------------------------------------------------------- utilities
__global__ void zero_kernel(float* __restrict__ p, int n) {
  int i = blockIdx.x * blockDim.x + threadIdx.x;
  if (i < n) p[i] = 0.0f;
}

// ---------------------------------------------------- layer-1 scalar scatter
// agg1[dst] += x[src]  (x is [N,1])
__global__ void scatter1_kernel(const float* __restrict__ x,
                                const int* __restrict__ edges,
                                float* __restrict__ agg1) {
  int e = blockIdx.x * blockDim.x + threadIdx.x;
  if (e < NE) {
    int s = edges[e];
    int d = edges[NE + e];
    atomicAdd(&agg1[d], x[s]);
  }
}

// ------------------------------------------------------------- layer-1 MLP
// h1 = relu( relu(z*W11 + b11) @ W12 + b12 ),  z = x + agg1 (eps = 0)
// one wave per 16x16 output tile; A (h0) synthesized on the fly (rank-1 input)
__global__ void l1_mlp_kernel(const float* __restrict__ x,
                              const float* __restrict__ agg1,
                              const float* __restrict__ W11,
                              const float* __restrict__ b11,
                              const float* __restrict__ W12,
                              const float* __restrict__ b12,
                              float* __restrict__ h1) {
  const int wave = threadIdx.x >> 5;    // N tile 0..7
  const int lane = threadIdx.x & 31;
  const int hi = lane >> 4;             // lane-half selects K pair
  const int lo = lane & 15;
  const int M0 = blockIdx.x * 16;
  const int N0 = wave * 16;

  const int row = M0 + lo;
  const float z = x[row] + agg1[row];

  v8f acc = {};
  #pragma unroll 4
  for (int k0 = 0; k0 < HID; k0 += 4) {
    const int ka = k0 + 2 * hi;
    v2f a, b;
    float t0 = fmaf(z, W11[ka],     b11[ka]);
    float t1 = fmaf(z, W11[ka + 1], b11[ka + 1]);
    a.x = t0 > 0.0f ? t0 : 0.0f;
    a.y = t1 > 0.0f ? t1 : 0.0f;
    b.x = W12[ka * HID + N0 + lo];
    b.y = W12[(ka + 1) * HID + N0 + lo];
    acc = __builtin_amdgcn_wmma_f32_16x16x4_f32(false, a, false, b,
                                                (short)0, acc, false, false);
  }
  const float bias = b12[N0 + lo];
  #pragma unroll
  for (int i = 0; i < 8; ++i) {
    float v = acc[i] + bias;
    v = v > 0.0f ? v : 0.0f;
    h1[(size_t)(M0 + i + 8 * hi) * HID + N0 + lo] = v;
  }
}

// ---------------------------------------------------- layer-2 vector scatter
// agg2[dst, :] += h1[src, :]; one wave per edge, float4 per lane
__global__ void scatter2_kernel(const float* __restrict__ h1,
                                const int* __restrict__ edges,
                                float* __restrict__ agg2) {
  const unsigned tid = blockIdx.x * blockDim.x + threadIdx.x;
  const int e = tid >> 5;
  const int lane = tid & 31;
  if (e < NE) {
    if (e + 2048 < NE) {
      __builtin_prefetch(&edges[e + 2048], 0, 1);
      __builtin_prefetch(&edges[NE + e + 2048], 0, 1);
    }
    const int s = edges[e];
    const int d = edges[NE + e];
    const float4 v = *(const float4*)(h1 + (size_t)s * HID + lane * 4);
    float* dp = agg2 + (size_t)d * HID + lane * 4;
    atomicAdd(dp + 0, v.x);
    atomicAdd(dp + 1, v.y);
    atomicAdd(dp + 2, v.z);
    atomicAdd(dp + 3, v.w);
  }
}

// --------------------------------------- layer-2 MLP + sum-pool (fused)
// X = h1 + agg2 ; T = relu(X@W21+b21) ; H = relu(T@W22+b22) ; gsum += colsum(H)
__global__ void l2_pool_kernel(const float* __restrict__ h1,
                               const float* __restrict__ agg2,
                               const float* __restrict__ W21,
                               const float* __restrict__ b21,
                               const float* __restrict__ W22,
                               const float* __restrict__ b22,
                               float* __restrict__ gsum) {
  __shared__ float Xt[16 * LDSS];
  __shared__ float Tt[16 * LDSS];
  const int wave = threadIdx.x >> 5;
  const int lane = threadIdx.x & 31;
  const int hi = lane >> 4;
  const int lo = lane & 15;
  const int M0 = blockIdx.x * 16;
  const int N0 = wave * 16;

  // cooperative staging of X tile (16 x 128) into padded LDS
  for (int i = threadIdx.x; i < 16 * HID; i += blockDim.x) {
    const int r = i >> 7, c = i & 127;
    const size_t gi = (size_t)(M0 + r) * HID + c;
    Xt[r * LDSS + c] = h1[gi] + agg2[gi];
  }
  __syncthreads();

  // GEMM1: T tile (this wave owns columns N0..N0+15)
  v8f acc = {};
  #pragma unroll 4
  for (int k0 = 0; k0 < HID; k0 += 4) {
    const int ka = k0 + 2 * hi;
    v2f a, b;
    a.x = Xt[lo * LDSS + ka];
    a.y = Xt[lo * LDSS + ka + 1];
    b.x = W21[ka * HID + N0 + lo];
    b.y = W21[(ka + 1) * HID + N0 + lo];
    acc = __builtin_amdgcn_wmma_f32_16x16x4_f32(false, a, false, b,
                                                (short)0, acc, false, false);
  }
  float bias = b21[N0 + lo];
  #pragma unroll
  for (int i = 0; i < 8; ++i) {
    float v = acc[i] + bias;
    Tt[(i + 8 * hi) * LDSS + N0 + lo] = v > 0.0f ? v : 0.0f;
  }
  __syncthreads();

  // GEMM2 + ReLU + column partial sum (pooling); h2 never hits global memory
  acc = (v8f){};
  #pragma unroll 4
  for (int k0 = 0; k0 < HID; k0 += 4) {
    const int ka = k0 + 2 * hi;
    v2f a, b;
    a.x = Tt[lo * LDSS + ka];
    a.y = Tt[lo * LDSS + ka + 1];
    b.x = W22[ka * HID + N0 + lo];
    b.y = W22[(ka + 1) * HID + N0 + lo];
    acc = __builtin_amdgcn_wmma_f32_16x16x4_f32(false, a, false, b,
                                                (short)0, acc, false, false);
  }
  bias = b22[N0 + lo];
  float s = 0.0f;
  #pragma unroll
  for (int i = 0; i < 8; ++i) {
    float v = acc[i] + bias;
    s += (v > 0.0f ? v : 0.0f);
  }
  atomicAdd(&gsum[N0 + lo], s);  // two lane-halves each add their 8-row partial
}

// ----------------------------------------------------------- final linear
__global__ void final_kernel(const float* __restrict__ gsum,
                             const float* __restrict__ Wf,
                             const float* __restrict__ bf,
                             float* __restrict__ out) {
  const int j = threadIdx.x;  // 128 threads
  float acc = bf[j];
  #pragma unroll 8
  for (int k = 0; k < HID; ++k) acc = fmaf(gsum[k], Wf[k * HID + j], acc);
  out[j] = acc;
}

extern "C" void kernel_launch(void* const* d_in, const int* in_sizes, int n_in,
                              void* d_out, int out_size, void* d_ws, size_t ws_size,
                              hipStream_t stream) {
  (void)in_sizes; (void)n_in; (void)out_size; (void)ws_size;
  const float* W11 = (const float*)d_in[6];
  const float* b11 = (const float*)d_in[7];
  const float* W12 = (const float*)d_in[8];
  const float* b12 = (const float*)d_in[9];
  const float* W21 = (const float*)d_in[10];
  const float* b21 = (const float*)d_in[11];
  const float* W22 = (const float*)d_in[12];
  const float* b22 = (const float*)d_in[13];
  const float* Wf  = (const float*)d_in[14];
  const float* bf  = (const float*)d_in[15];

  // workspace layout (floats): agg1 | agg2 | gsum  (zeroed together) | h1
  float* agg1 = (float*)d_ws;                 // NN
  float* agg2 = agg1 + NN;                    // NN*HID
  float* gsum = agg2 + (size_t)NN * HID;      // HID
  float* h1   = gsum + HID;                   // NN*HID (fully overwritten)
  const int nzero = NN + NN * HID + HID;      // 2,580,128

  float* out = (float*)d_out;

  for (int g = 0; g < 3; ++g) {
    const float* x = (const float*)d_in[2 * g];
    const int* edges = (const int*)d_in[2 * g + 1];

    zero_kernel<<<(nzero + 255) / 256, 256, 0, stream>>>(agg1, nzero);
    scatter1_kernel<<<(NE + 255) / 256, 256, 0, stream>>>(x, edges, agg1);
    l1_mlp_kernel<<<NN / 16, 256, 0, stream>>>(x, agg1, W11, b11, W12, b12, h1);
    scatter2_kernel<<<(NE * 32) / 256, 256, 0, stream>>>(h1, edges, agg2);
    l2_pool_kernel<<<NN / 16, 256, 0, stream>>>(h1, agg2, W21, b21, W22, b22, gsum);
    final_kernel<<<1, 128, 0, stream>>>(gsum, Wf, bf, out + g * HID);
  }
}